// ComplexEMA_25563645345978
// MI455X (gfx1250) — compile-verified
//
#include <hip/hip_runtime.h>

// ComplexEMA (diagonal complex SSM) on gfx1250 via chunked-scan GEMM formulation.
// y[b,d,:] = K_d(Toeplitz,16x16) @ x_chunks + A_re@H_re - A_im@H_im  (per 16-chunk tile)
// with inter-chunk states H scanned sequentially (cheap), all GEMMs in fp32 WMMA.

typedef float v8f __attribute__((ext_vector_type(8)));
typedef float v2f __attribute__((ext_vector_type(2)));

#define D_DIM 2048
#define N_MODES 16
#define B_DIM 2
#define L_DIM 4096
#define CG 128            // chunks per group (group spans CG*16 = 2048 timesteps)
#define NGROUP 2          // 2 groups cover L = 4096
#define MROW 17           // padded row stride for 16x16 coefficient matrices
#define UROW 136          // padded row stride for U/H arrays (128 chunks + pad)
#define XROW 17           // padded per-chunk stride for x/y staging

static __device__ __forceinline__ v8f wmma4(v2f a, v2f b, v8f c) {
  // D = A(16x4, f32) x B(4x16, f32) + C(16x16, f32)
  return __builtin_amdgcn_wmma_f32_16x16x4_f32(false, a, false, b, (short)0, c,
                                               false, false);
}

static __device__ __forceinline__ void cpow(float qr, float qi, int e,
                                            float& pr, float& pi) {
  pr = 1.f; pi = 0.f;
  for (int k = 0; k < e; ++k) {
    float t = pr * qr - pi * qi;
    pi = pr * qi + pi * qr;
    pr = t;
  }
}

__launch_bounds__(256, 2)
__global__ void cema_kernel(const float* __restrict__ x,
                            const float* __restrict__ alpha,
                            const float* __restrict__ delta,
                            const float* __restrict__ theta,
                            const float* __restrict__ gma,
                            const float* __restrict__ omga,
                            float* __restrict__ y_out,
                            float* __restrict__ h_out) {
  __shared__ float xs_s[B_DIM][CG * XROW];          // x staging; Y written in-place
  __shared__ float Ure_s[B_DIM][N_MODES * UROW];    // U then H (real)
  __shared__ float Uim_s[B_DIM][N_MODES * UROW];    // U then H (imag)
  __shared__ float Bre_s[N_MODES * MROW];           // B[n][sig] = p q^{15-sig} (re)
  __shared__ float Bim_s[N_MODES * MROW];
  __shared__ float Are_s[N_MODES * MROW];           // A[tau][n] = gam q^{tau+1} (re)
  __shared__ float AimN_s[N_MODES * MROW];          // -Im(gam q^{tau+1})
  __shared__ float Kmat_s[N_MODES * MROW];          // lower-tri Toeplitz + omega*I
  __shared__ float kern_s[N_MODES];
  __shared__ float cQr[16], cQi[16], cP[16], cGr[16], cGi[16];
  __shared__ float cCr[16], cCi[16], cTr[16], cTi[16];
  __shared__ float sOmega;

  const int tid = threadIdx.x;
  const int d = blockIdx.x;

  // ---------------- coefficient setup (per-mode scalars) ----------------
  if (tid < 16) {
    int n = tid;
    float a = alpha[d * 16 + n];
    float de = delta[d * 16 + n];
    float th = theta[d];
    float p = 1.f / (1.f + expf(-a));
    float dd = 1.f / (1.f + expf(-de));
    float base = (1.f / (1.f + expf(-th))) * (2.f * 3.14159265358979323846f / 16.f);
    float phi = (float)(n + 1) * base;
    float r = 1.f - p * dd;
    float sn, cs;
    sincosf(phi, &sn, &cs);
    float qr = r * cs, qi = r * sn;
    cQr[n] = qr; cQi[n] = qi;
    cP[n] = p;
    float gr = gma[(d * 16 + n) * 2 + 0] * 0.25f;
    float gi = gma[(d * 16 + n) * 2 + 1] * 0.25f;
    cGr[n] = gr; cGi[n] = gi;
    cCr[n] = p * gr; cCi[n] = p * gi;
    float tr, ti;
    cpow(qr, qi, 16, tr, ti);          // q^16 for the chunk scan
    cTr[n] = tr; cTi[n] = ti;
    if (n == 0) sOmega = omga[d];
  }
  __syncthreads();

  // ---------------- build 16x16 matrices in LDS ----------------
  {
    int r = tid >> 4, cc = tid & 15;
    // kernel terms: Re(c_cc * q_cc^r) -> stash in xs_s[0] (reloaded later)
    float qr = cQr[cc], qi = cQi[cc];
    float pr, pi;
    cpow(qr, qi, r, pr, pi);
    xs_s[0][r * MROW + cc] = cCr[cc] * pr - cCi[cc] * pi;
    // A[tau=r][n=cc] = gam_cc * q_cc^{r+1}
    float q1r = pr * qr - pi * qi;
    float q1i = pr * qi + pi * qr;
    Are_s[r * MROW + cc] = cGr[cc] * q1r - cGi[cc] * q1i;
    AimN_s[r * MROW + cc] = -(cGr[cc] * q1i + cGi[cc] * q1r);
    // B[n=r][sig=cc] = p_r * q_r^{15-cc}
    float br, bi;
    cpow(cQr[r], cQi[r], 15 - cc, br, bi);
    Bre_s[r * MROW + cc] = cP[r] * br;
    Bim_s[r * MROW + cc] = cP[r] * bi;
  }
  __syncthreads();
  if (tid < 16) {
    float s = 0.f;
    for (int n = 0; n < 16; ++n) s += xs_s[0][tid * MROW + n];
    kern_s[tid] = s;                   // kernel_d[m], m = tid
  }
  __syncthreads();
  {
    int r = tid >> 4, cc = tid & 15;
    float v = (r >= cc) ? kern_s[r - cc] : 0.f;
    if (r == cc) v += sOmega;          // fold omega*x into the diagonal
    Kmat_s[r * MROW + cc] = v;
  }

  // ---------------- main chunked pipeline ----------------
  const int wv = tid >> 5;             // wave id 0..7
  const int lane = tid & 31;
  const int lcol = lane & 15;          // WMMA: column (B/C/D) and row m (A)
  const int khalf = lane >> 4;         // 0: lanes 0-15, 1: lanes 16-31

  float hre = 0.f, him = 0.f;          // scan carry (meaningful for tid < 32)

  for (int g = 0; g < NGROUP; ++g) {
    __syncthreads();                   // matrices visible / prev group done
    // stage x for this group (coalesced float4 loads)
    for (int b2 = 0; b2 < B_DIM; ++b2) {
      const float4* xg = (const float4*)(x + ((size_t)b2 * D_DIM + d) * L_DIM +
                                         (size_t)g * (CG * 16));
      for (int i = tid; i < CG * 16 / 4; i += 256) {
        float4 v = xg[i];
        int t = i * 4;
        float* dst = &xs_s[b2][(t >> 4) * XROW + (t & 15)];
        dst[0] = v.x; dst[1] = v.y; dst[2] = v.z; dst[3] = v.w;
      }
    }
    __syncthreads();

    // Phase A: U = Bmat @ X  (16 tile-columns = 2 batches x 8 tiles; 2 per wave)
    #pragma unroll
    for (int u = 0; u < 2; ++u) {
      int asg = wv * 2 + u;
      int b2 = asg >> 3;
      int tile = asg & 7;
      const float* xb = xs_s[b2];
      int xbase = (tile * 16 + lcol) * XROW;
      v8f accR = {0.f, 0.f, 0.f, 0.f, 0.f, 0.f, 0.f, 0.f};
      v8f accI = {0.f, 0.f, 0.f, 0.f, 0.f, 0.f, 0.f, 0.f};
      #pragma unroll
      for (int kk = 0; kk < 4; ++kk) {
        int k0 = kk * 4 + khalf * 2;
        v2f bx; bx.x = xb[xbase + k0]; bx.y = xb[xbase + k0 + 1];
        v2f aR; aR.x = Bre_s[lcol * MROW + k0]; aR.y = Bre_s[lcol * MROW + k0 + 1];
        v2f aI; aI.x = Bim_s[lcol * MROW + k0]; aI.y = Bim_s[lcol * MROW + k0 + 1];
        accR = wmma4(aR, bx, accR);
        accI = wmma4(aI, bx, accI);
      }
      #pragma unroll
      for (int v = 0; v < 8; ++v) {
        int n = v + 8 * khalf;
        Ure_s[b2][n * UROW + tile * 16 + lcol] = accR[v];
        Uim_s[b2][n * UROW + tile * 16 + lcol] = accI[v];
      }
    }
    __syncthreads();

    // Scan: H_{j+1} = q^16 H_j + U_j ; overwrite U[j] with H_j (state BEFORE chunk j)
    if (tid < 32) {
      int b2 = tid >> 4, n = tid & 15;
      float qtr = cTr[n], qti = cTi[n];
      float* ur = &Ure_s[b2][n * UROW];
      float* ui = &Uim_s[b2][n * UROW];
      for (int j = 0; j < CG; ++j) {
        float tur = ur[j], tui = ui[j];
        ur[j] = hre; ui[j] = him;
        float nr = qtr * hre - qti * him + tur;
        float ni = qtr * him + qti * hre + tui;
        hre = nr; him = ni;
      }
    }
    __syncthreads();

    // Phase C: Y = Kmat@X + Are@Hre + AimN@Him ; write Y in-place over xs tile
    #pragma unroll
    for (int u = 0; u < 2; ++u) {
      int asg = wv * 2 + u;
      int b2 = asg >> 3;
      int tile = asg & 7;
      float* xb = xs_s[b2];
      int xbase = (tile * 16 + lcol) * XROW;
      v8f acc = {0.f, 0.f, 0.f, 0.f, 0.f, 0.f, 0.f, 0.f};
      #pragma unroll
      for (int kk = 0; kk < 4; ++kk) {
        int k0 = kk * 4 + khalf * 2;
        v2f bx; bx.x = xb[xbase + k0]; bx.y = xb[xbase + k0 + 1];
        v2f aK; aK.x = Kmat_s[lcol * MROW + k0]; aK.y = Kmat_s[lcol * MROW + k0 + 1];
        acc = wmma4(aK, bx, acc);
      }
      #pragma unroll
      for (int kk = 0; kk < 4; ++kk) {
        int k0 = kk * 4 + khalf * 2;
        v2f bh;
        bh.x = Ure_s[b2][k0 * UROW + tile * 16 + lcol];
        bh.y = Ure_s[b2][(k0 + 1) * UROW + tile * 16 + lcol];
        v2f aA; aA.x = Are_s[lcol * MROW + k0]; aA.y = Are_s[lcol * MROW + k0 + 1];
        acc = wmma4(aA, bh, acc);
      }
      #pragma unroll
      for (int kk = 0; kk < 4; ++kk) {
        int k0 = kk * 4 + khalf * 2;
        v2f bh;
        bh.x = Uim_s[b2][k0 * UROW + tile * 16 + lcol];
        bh.y = Uim_s[b2][(k0 + 1) * UROW + tile * 16 + lcol];
        v2f aA; aA.x = AimN_s[lcol * MROW + k0]; aA.y = AimN_s[lcol * MROW + k0 + 1];
        acc = wmma4(aA, bh, acc);
      }
      #pragma unroll
      for (int v = 0; v < 8; ++v) {
        xb[(tile * 16 + lcol) * XROW + v + 8 * khalf] = acc[v];
      }
    }
    __syncthreads();

    // stream Y out (coalesced float4 stores)
    for (int b2 = 0; b2 < B_DIM; ++b2) {
      float4* yg = (float4*)(y_out + ((size_t)b2 * D_DIM + d) * L_DIM +
                             (size_t)g * (CG * 16));
      for (int i = tid; i < CG * 16 / 4; i += 256) {
        int t = i * 4;
        const float* src = &xs_s[b2][(t >> 4) * XROW + (t & 15)];
        float4 v;
        v.x = src[0]; v.y = src[1]; v.z = src[2]; v.w = src[3];
        yg[i] = v;
      }
    }
  }

  // final states h_last -> (B, D, N, 2)
  if (tid < 32) {
    int b2 = tid >> 4, n = tid & 15;
    float* hp = h_out + (((size_t)b2 * D_DIM + d) * N_MODES + n) * 2;
    hp[0] = hre;
    hp[1] = him;
  }
}

extern "C" void kernel_launch(void* const* d_in, const int* in_sizes, int n_in,
                              void* d_out, int out_size, void* d_ws, size_t ws_size,
                              hipStream_t stream) {
  (void)in_sizes; (void)n_in; (void)out_size; (void)d_ws; (void)ws_size;
  const float* x = (const float*)d_in[0];
  const float* alpha = (const float*)d_in[1];
  const float* delta = (const float*)d_in[2];
  const float* theta = (const float*)d_in[3];
  const float* gma = (const float*)d_in[4];
  const float* omga = (const float*)d_in[5];
  float* y = (float*)d_out;
  float* h = (float*)d_out + (size_t)B_DIM * D_DIM * L_DIM;
  cema_kernel<<<dim3(D_DIM), dim3(256), 0, stream>>>(x, alpha, delta, theta,
                                                     gma, omga, y, h);
}